// GraphSage_36249523978328
// MI455X (gfx1250) — compile-verified
//
#include <hip/hip_runtime.h>
#include <hip/hip_bf16.h>

typedef float v2f __attribute__((ext_vector_type(2)));
typedef float v8f __attribute__((ext_vector_type(8)));

#define DIN 64
#define DH  128
#define DOUTF 64

// ---------------- zero fill (float4) ----------------
__global__ void zero_f4_kernel(float* __restrict__ p, long long n4) {
    long long i = (long long)blockIdx.x * blockDim.x + threadIdx.x;
    if (i < n4) {
        float4* q = (float4*)p;
        q[i] = make_float4(0.f, 0.f, 0.f, 0.f);
    }
}

// ---------------- degree count ----------------
__global__ void deg_count_kernel(const int* __restrict__ es, const int* __restrict__ ed,
                                 float* __restrict__ degU, float* __restrict__ degI, int E) {
    int e = blockIdx.x * blockDim.x + threadIdx.x;
    if (e < E) {
        __hip_atomic_fetch_add(&degU[es[e]], 1.0f, __ATOMIC_RELAXED, __HIP_MEMORY_SCOPE_AGENT);
        __hip_atomic_fetch_add(&degI[ed[e]], 1.0f, __ATOMIC_RELAXED, __HIP_MEMORY_SCOPE_AGENT);
    }
}

__global__ void recip_deg_kernel(float* __restrict__ deg, int n) {
    int i = blockIdx.x * blockDim.x + threadIdx.x;
    if (i < n) deg[i] = 1.0f / fmaxf(deg[i], 1.0f);
}

// ---------------- edge scatter: dst[sidx[e]] += src[gidx[e]] (float4 chunks) ----------------
__global__ void scatter_add_kernel(const float* __restrict__ src,
                                   const int* __restrict__ gidx,
                                   const int* __restrict__ sidx,
                                   float* __restrict__ dst,
                                   int E, int D) {
    const int chunks = D >> 2;
    int t = blockIdx.x * blockDim.x + threadIdx.x;
    int total = E * chunks;           // max 32M, fits int
    if (t >= total) return;
    int e = t / chunks;
    int c = (t - e * chunks) << 2;
    const float4 v = *(const float4*)(src + (size_t)gidx[e] * D + c);
    float* d = dst + (size_t)sidx[e] * D + c;
    __hip_atomic_fetch_add(d + 0, v.x, __ATOMIC_RELAXED, __HIP_MEMORY_SCOPE_AGENT);
    __hip_atomic_fetch_add(d + 1, v.y, __ATOMIC_RELAXED, __HIP_MEMORY_SCOPE_AGENT);
    __hip_atomic_fetch_add(d + 2, v.z, __ATOMIC_RELAXED, __HIP_MEMORY_SCOPE_AGENT);
    __hip_atomic_fetch_add(d + 3, v.w, __ATOMIC_RELAXED, __HIP_MEMORY_SCOPE_AGENT);
}

// ---------------- fused SAGE GEMM via fp32 WMMA ----------------
// out[n, 0:DOUT] = h[n,0:K] @ Ws + bs + (agg[n,0:K] * rdeg[n]) @ Wn   (+ optional ReLU)
// One wave per 16x16 output tile; K consumed 4/step by v_wmma_f32_16x16x4_f32.
__global__ __launch_bounds__(128) void sage_gemm_wmma_kernel(
        const float* __restrict__ h,
        const float* __restrict__ agg,
        const float* __restrict__ rdeg,
        const float* __restrict__ Ws,
        const float* __restrict__ Wn,
        const float* __restrict__ bs,
        float* __restrict__ out,
        int N, int K, int DOUT, int relu) {
    const int lane = threadIdx.x;                 // 0..31 (wave32)
    const int col_tiles = DOUT >> 4;
    const int w = blockIdx.x * 4 + threadIdx.y;   // global wave id
    const int row_tile = w / col_tiles;
    const int col_tile = w - row_tile * col_tiles;
    const int row_base = row_tile << 4;
    if (row_base >= N) return;                    // uniform across wave -> safe before WMMA

    const int m  = lane & 15;
    const int hi = lane >> 4;
    const int col = (col_tile << 4) + m;          // B/bias column for this lane (n = lane&15)

    int rowA = row_base + m;                      // A row held by this lane
    if (rowA >= N) rowA = N - 1;                  // select, no divergence
    const float rd = rdeg[rowA];
    const float* __restrict__ hrow = h   + (size_t)rowA * K;
    const float* __restrict__ arow = agg + (size_t)rowA * K;

    v8f c = {0.f, 0.f, 0.f, 0.f, 0.f, 0.f, 0.f, 0.f};

    for (int k0 = 0; k0 < K; k0 += 4) {
        const int ka = k0 + (hi << 1);            // this lane-half holds K = ka, ka+1
        v2f a_s, a_n, b_s, b_n;
        // A (16x4, fp32): v0 = K=2hi, v1 = K=2hi+1, rows striped over lanes 0..15
        a_s.x = hrow[ka];
        a_s.y = hrow[ka + 1];
        a_n.x = arow[ka] * rd;
        a_n.y = arow[ka + 1] * rd;
        // B (4x16, fp32): columns striped over lanes, rows ka, ka+1 in the two regs
        b_s.x = Ws[(size_t)ka * DOUT + col];
        b_s.y = Ws[(size_t)(ka + 1) * DOUT + col];
        b_n.x = Wn[(size_t)ka * DOUT + col];
        b_n.y = Wn[(size_t)(ka + 1) * DOUT + col];
        // D = A*B + C ; chain both the self path and the neighbor path into c
        c = __builtin_amdgcn_wmma_f32_16x16x4_f32(false, a_s, false, b_s,
                                                  (short)0, c, false, false);
        c = __builtin_amdgcn_wmma_f32_16x16x4_f32(false, a_n, false, b_n,
                                                  (short)0, c, false, false);
    }

    const float bias = bs[col];
    // C/D layout: vgpr r, lane 16*hi + n -> D[row_base + r + 8*hi][col]
    #pragma unroll
    for (int r = 0; r < 8; ++r) {
        const int row = row_base + r + (hi << 3);
        if (row < N) {
            float v = c[r] + bias;
            if (relu) v = fmaxf(v, 0.0f);
            out[(size_t)row * DOUT + col] = v;
        }
    }
}

// ---------------- host-side helpers ----------------
static inline void launch_zero(float* p, long long nfloats, hipStream_t s) {
    long long n4 = nfloats >> 2;   // all our sizes are multiples of 4
    int blk = 256;
    long long grid = (n4 + blk - 1) / blk;
    zero_f4_kernel<<<(dim3)(unsigned)grid, blk, 0, s>>>(p, n4);
}

static inline void launch_scatter(const float* src, const int* gidx, const int* sidx,
                                  float* dst, int E, int D, hipStream_t s) {
    int total = E * (D >> 2);
    int blk = 256;
    int grid = (total + blk - 1) / blk;
    scatter_add_kernel<<<grid, blk, 0, s>>>(src, gidx, sidx, dst, E, D);
}

static inline void launch_gemm(const float* h, const float* agg, const float* rdeg,
                               const float* Ws, const float* Wn, const float* bs,
                               float* out, int N, int K, int DOUT, int relu, hipStream_t s) {
    int tiles = ((N + 15) >> 4) * (DOUT >> 4);
    int grid = (tiles + 3) >> 2;                  // 4 waves / block
    sage_gemm_wmma_kernel<<<grid, dim3(32, 4), 0, s>>>(h, agg, rdeg, Ws, Wn, bs,
                                                       out, N, K, DOUT, relu);
}

extern "C" void kernel_launch(void* const* d_in, const int* in_sizes, int n_in,
                              void* d_out, int out_size, void* d_ws, size_t ws_size,
                              hipStream_t stream) {
    const float* feat_u   = (const float*)d_in[0];
    const float* feat_i   = (const float*)d_in[1];
    const int*   edge_src = (const int*)d_in[2];
    const int*   edge_dst = (const int*)d_in[3];
    const float* Wn0 = (const float*)d_in[4];
    const float* Ws0 = (const float*)d_in[5];
    const float* bs0 = (const float*)d_in[6];
    const float* Wn1 = (const float*)d_in[7];
    const float* Ws1 = (const float*)d_in[8];
    const float* bs1 = (const float*)d_in[9];
    const float* Wn2 = (const float*)d_in[10];
    const float* Ws2 = (const float*)d_in[11];
    const float* bs2 = (const float*)d_in[12];

    const int NU = in_sizes[0] / DIN;
    const int NI = in_sizes[1] / DIN;
    const int E  = in_sizes[2];

    // workspace carve-up (floats)
    float* ws = (float*)d_ws;
    size_t o = 0;
    float* hu0  = ws + o; o += (size_t)NU * DH;
    float* hu1  = ws + o; o += (size_t)NU * DH;
    float* hi0  = ws + o; o += (size_t)NI * DH;
    float* hi1  = ws + o; o += (size_t)NI * DH;
    float* aggU = ws + o; o += (size_t)NU * DH;
    float* aggI = ws + o; o += (size_t)NI * DH;
    float* rdegU = ws + o; o += (size_t)NU;      // rdegU and rdegI contiguous
    float* rdegI = ws + o; o += (size_t)NI;

    // ---- degrees: count, then reciprocal of max(deg,1) ----
    launch_zero(rdegU, (long long)(NU + NI), stream);
    {
        int blk = 256, grid = (E + blk - 1) / blk;
        deg_count_kernel<<<grid, blk, 0, stream>>>(edge_src, edge_dst, rdegU, rdegI, E);
        int n = NU + NI;
        recip_deg_kernel<<<(n + blk - 1) / blk, blk, 0, stream>>>(rdegU, n);
    }

    // ---- layer 0 : K = DIN(64) -> DH(128), ReLU fused into outputs ----
    launch_zero(aggU, (long long)NU * DIN, stream);
    launch_zero(aggI, (long long)NI * DIN, stream);
    launch_scatter(feat_i, edge_dst, edge_src, aggU, E, DIN, stream); // items -> users
    launch_scatter(feat_u, edge_src, edge_dst, aggI, E, DIN, stream); // users -> items
    launch_gemm(feat_u, aggU, rdegU, Ws0, Wn0, bs0, hu0, NU, DIN, DH, 1, stream);
    launch_gemm(feat_i, aggI, rdegI, Ws0, Wn0, bs0, hi0, NI, DIN, DH, 1, stream);

    // ---- layer 1 : K = DH -> DH, ReLU fused ----
    launch_zero(aggU, (long long)NU * DH, stream);
    launch_zero(aggI, (long long)NI * DH, stream);
    launch_scatter(hi0, edge_dst, edge_src, aggU, E, DH, stream);
    launch_scatter(hu0, edge_src, edge_dst, aggI, E, DH, stream);
    launch_gemm(hu0, aggU, rdegU, Ws1, Wn1, bs1, hu1, NU, DH, DH, 1, stream);
    launch_gemm(hi0, aggI, rdegI, Ws1, Wn1, bs1, hi1, NI, DH, DH, 1, stream);

    // ---- layer 2 : only the item side is needed (output = h_i), no ReLU ----
    launch_zero(aggI, (long long)NI * DH, stream);
    launch_scatter(hu1, edge_src, edge_dst, aggI, E, DH, stream);
    launch_gemm(hi1, aggI, rdegI, Ws2, Wn2, bs2, (float*)d_out, NI, DH, DOUTF, 0, stream);
}